// SpikingReadoutLayer_7189775254185
// MI455X (gfx1250) — compile-verified
//
#include <hip/hip_runtime.h>
#include <cmath>

// Problem constants (match the reference)
#define ALPHA_F 0.95f
#define BETA_F  0.9f
constexpr int H = 256;      // hidden
constexpr int O = 16;       // outputs
constexpr int T = 2000;     // time steps
constexpr int TPB = 256;    // 8 waves (wave32)
constexpr int WAVES = 8;
constexpr int CHUNK = T / WAVES;   // 250
constexpr int TILES = T / 16;      // 125 row-tiles of 16

typedef __attribute__((ext_vector_type(2))) float    v2f;
typedef __attribute__((ext_vector_type(8))) float    v8f;
typedef __attribute__((ext_vector_type(4))) uint32_t v4u;
typedef __attribute__((ext_vector_type(8))) int      v8i;
typedef __attribute__((ext_vector_type(4))) int      v4i;

__global__ __launch_bounds__(TPB, 1)
void spiking_readout_fused(const float* __restrict__ in,    // [B, T, H]
                           const float* __restrict__ w,     // [H, O]
                           float* __restrict__ out,         // [B, T+1, O]
                           float an, float bn, float gn)    // alpha^250, beta^250, (an-bn)/(a-b)
{
    // On-chip staging: h2[b] (125 KB) + weights (16 KB) + scan state.
    __shared__ float h2s[T * O];           // 128000 B
    __shared__ float wlds[H * O];          // 16384 B
    __shared__ float segF[WAVES * O];
    __shared__ float segO[WAVES * O];
    __shared__ float iniF[WAVES * O];
    __shared__ float iniO[WAVES * O];

    const int tid  = threadIdx.x;
    const int b    = blockIdx.x;
    const int lane = tid & 31;
    const int wave = tid >> 5;
    const int m    = lane & 15;   // A row / B col / D col
    const int hi   = lane >> 4;   // half-wave select

    // ---- stage weights [H,O] into LDS via the Tensor Data Mover ----
    // D# per ISA ch.8: flat 1-row tensor of 4096 f32 elements.
#if defined(__gfx1250__) && __has_builtin(__builtin_amdgcn_tensor_load_to_lds)
    if (tid < 32) {                        // one wave issues the DMA (EXEC ignored by TDM)
        const uint64_t ga = (uint64_t)(uintptr_t)w;
        const uint32_t la = (uint32_t)(uintptr_t)&wlds[0];
        v4u g0;
        g0[0] = 1u;                                     // count=1, user descriptor
        g0[1] = la;                                     // lds_addr
        g0[2] = (uint32_t)ga;                           // global_addr[31:0]
        g0[3] = (uint32_t)(ga >> 32) | (2u << 30);      // global_addr[56:32] | type=2
        v8i g1 = {0, 0, 0, 0, 0, 0, 0, 0};
        g1[0] = (2 << 16);                              // wg_mask=0, data_size=2 (4 bytes)
        g1[1] = (int)(4096u << 16);                     // tensor_dim0[15:0] = 4096 @ bits 63:48
        g1[2] = (1 << 16);                              // tensor_dim1 = 1 @ bits 95:80
        g1[3] = (int)(4096u << 16);                     // tile_dim0 = 4096 @ bits 127:112
        g1[5] = 4096;                                   // tensor_dim0_stride[31:0] @ bits 191:160
        v4i g2 = {0, 0, 0, 0};
        v4i g3 = {0, 0, 0, 0};
        v8i g4 = {0, 0, 0, 0, 0, 0, 0, 0};              // 6-arg toolchain form
        __builtin_amdgcn_tensor_load_to_lds(g0, g1, g2, g3, g4, 0);
#if __has_builtin(__builtin_amdgcn_s_wait_tensorcnt)
        __builtin_amdgcn_s_wait_tensorcnt(0);
#else
        asm volatile("s_wait_tensorcnt 0x0" ::: "memory");
#endif
    }
#else
    for (int i = tid; i < H * O; i += TPB) wlds[i] = w[i];
#endif
    __syncthreads();

    const float* inb = in + (size_t)b * T * H;

    // ---- Phase 1: GEMM tiles -> LDS via V_WMMA_F32_16X16X4_F32 ----
    // 4 tiles per wave-iteration: one B fragment feeds 4 independent WMMA
    // accumulator chains (amortizes ds loads, removes C->D RAW stalls).
    for (int g = wave; g * 4 < TILES; g += WAVES) {
        int tt = g * 4;
        if (tt + 4 <= TILES) {
            v8f a0 = {0.f,0.f,0.f,0.f,0.f,0.f,0.f,0.f};
            v8f a1 = a0, a2v = a0, a3 = a0;
            const float* abase = inb + (size_t)(tt * 16 + m) * H + 2 * hi;
            #pragma unroll 2
            for (int kk = 0; kk < H / 4; ++kk) {
                const int k0 = kk * 4;
                v2f bf;
                bf.x = wlds[(k0 + 2 * hi)     * O + m];
                bf.y = wlds[(k0 + 2 * hi + 1) * O + m];
                v2f f0 = *(const v2f*)(abase + k0);
                v2f f1 = *(const v2f*)(abase + 16 * H + k0);
                v2f f2 = *(const v2f*)(abase + 32 * H + k0);
                v2f f3 = *(const v2f*)(abase + 48 * H + k0);
                a0  = __builtin_amdgcn_wmma_f32_16x16x4_f32(false, f0, false, bf, (short)0, a0,  false, false);
                a1  = __builtin_amdgcn_wmma_f32_16x16x4_f32(false, f1, false, bf, (short)0, a1,  false, false);
                a2v = __builtin_amdgcn_wmma_f32_16x16x4_f32(false, f2, false, bf, (short)0, a2v, false, false);
                a3  = __builtin_amdgcn_wmma_f32_16x16x4_f32(false, f3, false, bf, (short)0, a3,  false, false);
            }
            // D layout: VGPR v, lane -> (M = v + 8*hi, N = m)
            const int base = (tt * 16 + 8 * hi) * O + m;
            #pragma unroll
            for (int v = 0; v < 8; ++v) {
                h2s[base +            v * O] = a0[v];
                h2s[base + 16 * O  +  v * O] = a1[v];
                h2s[base + 32 * O  +  v * O] = a2v[v];
                h2s[base + 48 * O  +  v * O] = a3[v];
            }
        } else {
            // ragged tail (TILES % 4): one tile at a time
            for (; tt < TILES; ++tt) {
                v8f acc = {0.f,0.f,0.f,0.f,0.f,0.f,0.f,0.f};
                const float* arow = inb + (size_t)(tt * 16 + m) * H + 2 * hi;
                for (int kk = 0; kk < H / 4; ++kk) {
                    const int k0 = kk * 4;
                    v2f af = *(const v2f*)(arow + k0);
                    v2f bf;
                    bf.x = wlds[(k0 + 2 * hi)     * O + m];
                    bf.y = wlds[(k0 + 2 * hi + 1) * O + m];
                    acc = __builtin_amdgcn_wmma_f32_16x16x4_f32(false, af, false, bf, (short)0, acc, false, false);
                }
                const int base = (tt * 16 + 8 * hi) * O + m;
                #pragma unroll
                for (int v = 0; v < 8; ++v) h2s[base + v * O] = acc[v];
            }
        }
    }
    __syncthreads();

    // ---- Phase 2a: per-chunk local scan from zero state (128 threads) ----
    const int seg = tid >> 4;   // chunk id 0..7
    const int oc  = tid & 15;   // output channel
    if (tid < WAVES * O) {
        float flt = 0.f, oo = 0.f;
        const int tbase = seg * CHUNK;
        for (int i = 0; i < CHUNK; ++i) {
            const float h  = h2s[(tbase + i) * O + oc];
            const float no = BETA_F * oo + flt;     // uses OLD flt
            const float nf = ALPHA_F * flt + h;
            flt = nf; oo = no;
        }
        segF[tid] = flt;
        segO[tid] = oo;
    }
    __syncthreads();

    // ---- Phase 2b: 8-segment affine prefix (16 threads), A^n = [[an,0],[gn,bn]] ----
    if (tid < O) {
        float cf = 0.f, co = 0.f;
        for (int j = 0; j < WAVES; ++j) {
            iniF[j * O + tid] = cf;
            iniO[j * O + tid] = co;
            const float sf = segF[j * O + tid];
            const float so = segO[j * O + tid];
            const float nf = an * cf + sf;
            const float no = gn * cf + bn * co + so;
            cf = nf; co = no;
        }
        out[(size_t)b * (T + 1) * O + tid] = 0.f;   // out_rec[:,0,:] = 0
    }
    __syncthreads();

    // ---- Phase 2c: re-scan each chunk with correct init, stream outputs ----
    if (tid < WAVES * O) {
        float flt = iniF[tid], oo = iniO[tid];
        const int tbase = seg * CHUNK;
        float* op = out + ((size_t)b * (T + 1) + tbase + 1) * O + oc;
        for (int i = 0; i < CHUNK; ++i) {
            const float h  = h2s[(tbase + i) * O + oc];
            const float no = BETA_F * oo + flt;     // uses OLD flt
            const float nf = ALPHA_F * flt + h;
            op[(size_t)i * O] = no;                 // 16 lanes -> contiguous 64B store
            flt = nf; oo = no;
        }
    }
}

extern "C" void kernel_launch(void* const* d_in, const int* in_sizes, int n_in,
                              void* d_out, int out_size, void* d_ws, size_t ws_size,
                              hipStream_t stream) {
    const float* inputs  = (const float*)d_in[0];  // [B, T, H]
    const float* weights = (const float*)d_in[1];  // [H, O]
    float* out = (float*)d_out;                    // [B, T+1, O]

    const int batches = in_sizes[0] / (T * H);     // 128

    // Segment-jump affine coefficients for the 250-step chunks:
    // per-step transition [[alpha,0],[1,beta]]; its 250th power is
    // [[a^n, 0], [g, b^n]] with g = (a^n - b^n)/(a - b).
    const double a = 0.95, bt = 0.9;
    const double an_d = std::pow(a, (double)CHUNK);
    const double bn_d = std::pow(bt, (double)CHUNK);
    const double gn_d = (an_d - bn_d) / (a - bt);

    spiking_readout_fused<<<batches, TPB, 0, stream>>>(
        inputs, weights, out, (float)an_d, (float)bn_d, (float)gn_d);
}